// SphericalBessel_90168543412737
// MI455X (gfx1250) — compile-verified
//
#include <hip/hip_runtime.h>
#include <math.h>

// Spherical Bessel j_l(r*k) for l = floor(sqrt(m)), m = 0..99, matching the
// JAX reference: f64 upward recurrence (x >= l) + f64 Miller downward
// recurrence from l_start = 18 (x < l), selected on f32 x32 < l, output f32.
//
// MI455X reasoning: output is ~105 MB (4.5 us at 23.3 TB/s) but per-element
// f64 work (~150-200 VALU ops incl. sincos) dominates -> f64-VALU bound.
// Optimizations: single sincos + single reciprocal shared by both branches,
// sin(x)/x reused as the Miller normalizer, wave-uniform __any() skipping,
// fully unrolled fma recurrences, predicated per-lane capture of the one
// needed order, LDS-staged k table via gfx1250 async global->LDS load,
// coalesced stores (mode index fastest).

#define LMAX    9
#define KMODES  100
#define BLOCK   256
#define LSTART  18                    // LMAX + int(sqrt(10*LMAX)) = 9 + 9
#define S2OPI   0.7978845608028654    // sqrt(2/pi), double (Python float)

// Address-space-qualified int typedefs for the async-to-LDS builtin
// (clang reports its parameters as AS1/AS3 `int *`).
typedef __attribute__((address_space(1))) int as1_int;
typedef __attribute__((address_space(3))) int as3_int;

#if defined(__HIP_DEVICE_COMPILE__) && \
    __has_builtin(__builtin_amdgcn_global_load_async_to_lds_b32) && \
    __has_builtin(__builtin_amdgcn_s_wait_asynccnt)
#define USE_ASYNC_LDS 1
#else
#define USE_ASYNC_LDS 0
#endif

__global__ __launch_bounds__(BLOCK)
void SphericalBessel_90168543412737_kernel(const float* __restrict__ r,
                                           const float* __restrict__ kmodes,
                                           float* __restrict__ out,
                                           int total /* N*KMODES */) {
    __shared__ float sk[KMODES];

    const int tid = threadIdx.x;
#if USE_ASYNC_LDS
    if (tid < KMODES) {
        // gfx1250 async DMA: global -> LDS, tracked by ASYNCcnt.
        __builtin_amdgcn_global_load_async_to_lds_b32(
            (as1_int*)(kmodes + tid), (as3_int*)(&sk[tid]), 0, 0);
    }
    __builtin_amdgcn_s_wait_asynccnt(0);
#else
    if (tid < KMODES) sk[tid] = kmodes[tid];
#endif
    __syncthreads();

    const int gi = blockIdx.x * BLOCK + tid;
    if (gi >= total) return;

    const int n = gi / KMODES;        // mulhi trick (constant divisor)
    const int m = gi - n * KMODES;

    const float rf = r[n];            // broadcast within groups of 100 lanes
    const float kf = sk[m];
    const float lf = floorf(sqrtf((float)m));   // l for this (l,m) mode
    const int   l  = (int)lf;

    // ---- shared precomputation (used by both recurrences) ----
    const double x    = (double)rf * (double)kf;
    const double invx = 1.0 / x;
    double s, c;
    sincos(x, &s, &c);
    const double sincx = s * invx;    // sin(x)/x == j0(x) == reference sinc

    const float x32   = rf * kf;      // reference compares in f32
    const bool  useDn = (x32 < lf);

    double yu = 0.0, yd = 0.0;

    // ---- upward recurrence (reference _bessel_up), stable for x >= l ----
    if (__any(!useDn)) {
        double a = sincx;             // j0
        double b = (a - c) * invx;    // j1 = j0/x - cos(x)/x
        yu = (l == 0) ? a : b;        // covers l==0 and l==1
        #pragma unroll
        for (int i = 1; i < LMAX; ++i) {
            double jn = (double)(2 * i + 1) * invx * b - a;  // order i+1
            if (l == i + 1) yu = jn;  // predicated capture, no divergence
            a = b; b = jn;
        }
        if (fabs(x) < 1e-30) yu = 0.0;
        yu *= S2OPI * (double)kf;
    }

    // ---- downward (Miller) recurrence (reference _bessel_down), x < l ----
    if (__any(useDn)) {
        double j2 = 0.0, j1 = 1.0, j0 = 1.0;
        #pragma unroll
        for (int i = LSTART; i >= 1; --i) {
            j0 = (double)(2 * i + 1) * invx * j1 - j2;       // order i-1
            if (i - 1 < LMAX + 1) {
                if (l == i - 1) yd = j0;
            }
            j2 = j1; j1 = j0;
        }
        yd = yd / j0 * sincx;         // normalize by true j0 = sin(x)/x
        if (x < 1e-20) yd = (l > 0) ? 0.0 : 1.0;
        yd *= S2OPI * (double)kf;
    }

    out[gi] = useDn ? (float)yd : (float)yu;   // coalesced f32 store
}

extern "C" void kernel_launch(void* const* d_in, const int* in_sizes, int n_in,
                              void* d_out, int out_size, void* d_ws, size_t ws_size,
                              hipStream_t stream) {
    const float* r = (const float*)d_in[0];   // [N] f32
    const float* k = (const float*)d_in[1];   // [KMODES] f32
    float* out = (float*)d_out;               // [N, KMODES] f32

    const int total  = out_size;              // N * KMODES
    const int blocks = (total + BLOCK - 1) / BLOCK;
    SphericalBessel_90168543412737_kernel<<<blocks, BLOCK, 0, stream>>>(r, k, out, total);
}